// resize_module_cv2_58325655879845
// MI455X (gfx1250) — compile-verified
//
#include <hip/hip_runtime.h>
#include <hip/hip_bf16.h>

typedef __attribute__((ext_vector_type(2))) float v2f;
typedef __attribute__((ext_vector_type(8))) float v8f;

namespace {
constexpr int kH1 = 512, kW1 = 512, kH2 = 224, kW2 = 224;
constexpr int kPlanes = 64 * 3;
constexpr int kTilesH = kH2 / 16;            // 14
constexpr int kTilesW = kW2 / 16;            // 14
constexpr float kInvScale = 512.0f / 224.0f; // in/out = 16/7 (same for H and W)
constexpr float kKernelScale = kInvScale;    // antialias widens the triangle
constexpr int kBand = 40;    // >= max taps (<=39) spanned by a 16-wide tile
constexpr int kXStride = 41; // odd LDS strides -> conflict-free WMMA frag reads
constexpr int kWStride = 41;
constexpr int kWTabTile = 16 * kBand;                    // 640 floats per tile
constexpr size_t kWTabBytes = (size_t)kTilesH * kWTabTile * sizeof(float);
} // namespace

// Band start for an output tile (same math in both kernels).
__device__ __forceinline__ int band_start(int o0) {
  return (int)ceilf(((float)o0 + 0.5f) * kInvScale - 0.5f - kKernelScale);
}

// Normalized antialiased triangle weights for one output row.
__device__ __forceinline__ void weight_row(int o, int base, float* wrow) {
  const float sf = ((float)o + 0.5f) * kInvScale - 0.5f;
  float s = 0.0f;
#pragma unroll
  for (int k = 0; k < kBand; ++k) {
    const int u = base + k;
    const float x = fabsf(sf - (float)u) * (1.0f / kKernelScale);
    float w = fmaxf(0.0f, 1.0f - x);
    if (u < 0 || u >= kH1) w = 0.0f;         // kH1 == kW1 == 512
    wrow[k] = w;
    s += w;
  }
  const float inv = 1.0f / s;                // nearest tap always in range -> s>0
#pragma unroll
  for (int k = 0; k < kBand; ++k) wrow[k] *= inv;
}

// One shared weight table (H and W scales are identical): wtab[t][r][k].
__global__ __launch_bounds__(32)
void build_weight_table(float* __restrict__ wtab) {
  const int t = blockIdx.x;                  // 0..13
  const int r = (int)threadIdx.x;
  if (r >= 16) return;
  const int base = band_start(t * 16);
  float wrow[kBand];
  weight_row(t * 16 + r, base, wrow);
#pragma unroll
  for (int k = 0; k < kBand; ++k) wtab[t * kWTabTile + r * kBand + k] = wrow[k];
}

__global__ __launch_bounds__(32)
void resize_bilinear_aa_wmma(const float* __restrict__ in,
                             float* __restrict__ out,
                             const float* __restrict__ wtab) {
  __shared__ float Xs[48 * kXStride];   // input band (rows 40..47 unused pad)
  __shared__ float Whs[16 * kWStride];  // vertical weights  [i][slot]
  __shared__ float Wws[16 * kWStride];  // horizontal weights[j][slot]
  __shared__ float Ms[kBand * 16];      // intermediate M[u][j]

  const int tile = blockIdx.x;
  const int tw = tile % kTilesW;
  const int th = (tile / kTilesW) % kTilesH;
  const int plane = tile / (kTilesW * kTilesH);

  const int i0 = th * 16;
  const int j0 = tw * 16;
  const int u_start = band_start(i0);
  const int v_start = band_start(j0);

  const int lane = (int)threadIdx.x;
  const int r = lane & 15;
  const int half = lane >> 4;
  const int r8 = lane >> 3;                  // 0..3
  const int c8 = lane & 7;                   // 0..7

  // ---- stage weights into LDS (precomputed table; VALU fallback if no ws) ----
  if (wtab) {
    const float* __restrict__ whp = wtab + th * kWTabTile;
    const float* __restrict__ wwp = wtab + tw * kWTabTile;
    // cols 0..31: unrolled, immediate offsets off one base per array
#pragma unroll
    for (int rr = 0; rr < 16; ++rr) {
      Whs[rr * kWStride + lane] = whp[rr * kBand + lane];
      Wws[rr * kWStride + lane] = wwp[rr * kBand + lane];
    }
    // cols 32..39: shift/mask lane split, 4 unrolled iterations
#pragma unroll
    for (int it = 0; it < 4; ++it) {
      const int rr = it * 4 + r8;
      const int cc = 32 + c8;
      Whs[rr * kWStride + cc] = whp[rr * kBand + cc];
      Wws[rr * kWStride + cc] = wwp[rr * kBand + cc];
    }
  } else {
    const bool horiz = (half != 0);          // lanes 16-31: horizontal rows
    float wrow[kBand];
    weight_row((horiz ? j0 : i0) + r, horiz ? v_start : u_start, wrow);
    float* dst = horiz ? Wws : Whs;
#pragma unroll
    for (int k = 0; k < kBand; ++k) dst[r * kWStride + k] = wrow[k];
  }

  // ---- load 40x40 input band into LDS ----
  {
    const float* __restrict__ src = in + (size_t)plane * (kH1 * kW1);
    const bool interior = (u_start >= 0) & (u_start + kBand <= kH1) &
                          (v_start >= 0) & (v_start + kBand <= kW1);
    if (interior) {
      // Fast path (12/14 of tiles each way): unclamped, fully unrolled ->
      // each load/store is one instruction with an immediate offset.
      const float* __restrict__ srow = src + (size_t)u_start * kW1 + v_start;
      const float* __restrict__ s1 = srow + lane;
      float* __restrict__ x1 = &Xs[lane];
#pragma unroll
      for (int rr = 0; rr < kBand; ++rr)        // cols 0..31, 40 rows
        x1[rr * kXStride] = s1[(size_t)rr * kW1];
      const float* __restrict__ s2 = srow + (size_t)r8 * kW1 + 32 + c8;
      float* __restrict__ x2 = &Xs[r8 * kXStride + 32 + c8];
#pragma unroll
      for (int it = 0; it < 10; ++it)           // cols 32..39, 4 rows/iter
        x2[it * 4 * kXStride] = s2[(size_t)it * 4 * kW1];
    } else {
      // Edge tiles: clamped generic loop (values under zero weights are
      // irrelevant but must come from valid memory).
      for (int idx = lane; idx < kBand * kBand; idx += 32) {
        const int rr = idx / kBand;
        const int cc = idx - rr * kBand;
        const int u = min(max(u_start + rr, 0), kH1 - 1);
        const int v = min(max(v_start + cc, 0), kW1 - 1);
        Xs[rr * kXStride + cc] = src[u * kW1 + v];
      }
    }
  }
  __syncthreads();

  // ---- stage 1: M[u][j] = sum_v X[u][v] * Ww[j][v] ----
  // 3 row tiles (band padded to 48) x 10 K-steps of v_wmma_f32_16x16x4_f32.
  // Pad rows 40..47 read undefined LDS, but WMMA D rows are independent and
  // those rows are never stored to Ms.
#pragma unroll
  for (int t = 0; t < 3; ++t) {
    v8f acc = {0.f, 0.f, 0.f, 0.f, 0.f, 0.f, 0.f, 0.f};
#pragma unroll
    for (int kc = 0; kc < kBand / 4; ++kc) {
      const int vk = kc * 4 + 2 * half;
      v2f a, b;
      a.x = Xs[(t * 16 + r) * kXStride + vk];       // A[M=r][K]
      a.y = Xs[(t * 16 + r) * kXStride + vk + 1];
      b.x = Wws[r * kWStride + vk];                 // B[K][N=r] = Ww[r][v]
      b.y = Wws[r * kWStride + vk + 1];
      acc = __builtin_amdgcn_wmma_f32_16x16x4_f32(false, a, false, b,
                                                  (short)0, acc, false, false);
    }
    // D row -> M row; only t==2/upper-half rows (40..47) are discarded.
    if (t < 2) {
#pragma unroll
      for (int p = 0; p < 8; ++p) Ms[(t * 16 + p + 8 * half) * 16 + r] = acc[p];
    } else if (half == 0) {
#pragma unroll
      for (int p = 0; p < 8; ++p) Ms[(32 + p) * 16 + r] = acc[p];
    }
  }
  __syncthreads();

  // ---- stage 2: Out[i][j] = sum_u Wh[i][u] * M[u][j], 10 K-steps ----
  {
    v8f acc = {0.f, 0.f, 0.f, 0.f, 0.f, 0.f, 0.f, 0.f};
#pragma unroll
    for (int kc = 0; kc < kBand / 4; ++kc) {
      const int uk = kc * 4 + 2 * half;
      v2f a, b;
      a.x = Whs[r * kWStride + uk];                 // A[M=i=r][K=u]
      a.y = Whs[r * kWStride + uk + 1];
      b.x = Ms[uk * 16 + r];                        // B[K=u][N=j=r]
      b.y = Ms[(uk + 1) * 16 + r];
      acc = __builtin_amdgcn_wmma_f32_16x16x4_f32(false, a, false, b,
                                                  (short)0, acc, false, false);
    }
    float* __restrict__ dst = out + (size_t)plane * (kH2 * kW2);
#pragma unroll
    for (int p = 0; p < 8; ++p) {
      const int m = i0 + p + 8 * half;
      dst[m * kW2 + (j0 + r)] = acc[p];             // 16 lanes -> 64B rows
    }
  }
}

extern "C" void kernel_launch(void* const* d_in, const int* in_sizes, int n_in,
                              void* d_out, int out_size, void* d_ws,
                              size_t ws_size, hipStream_t stream) {
  (void)in_sizes; (void)n_in; (void)out_size;
  const float* x = (const float*)d_in[0];
  float* y = (float*)d_out;
  float* wtab = (ws_size >= kWTabBytes) ? (float*)d_ws : nullptr;
  if (wtab) build_weight_table<<<kTilesH, 32, 0, stream>>>(wtab);
  const int blocks = kPlanes * kTilesH * kTilesW;   // 192 * 14 * 14 = 37632
  resize_bilinear_aa_wmma<<<blocks, 32, 0, stream>>>(x, y, wtab);
}